// XGate_71150428226262
// MI455X (gfx1250) — compile-verified
//
#include <hip/hip_runtime.h>

// XGate (qudit shift) on a 3^16 f32 state vector, site INDEX=5, shift S=1.
// M = roll(eye(3),1) is a permutation => out[e] = x[e + delta(i)],
//   i = (e / RIGHT) % 3, delta(0)=+2*RIGHT, delta(1)=delta(2)=-RIGHT.
// Bit-identical to the reference f32 contraction (0*x and 1*x are exact).
//
// Roofline: 344 MB @ 23.3 TB/s -> ~14.8 us floor, 0 FLOPs. Optimal kernel is
// a streaming copy. delta mod 4 in {2,3} => source and destination can never
// both be 16B/8B aligned, so dword-per-lane is the correct granule (fully
// coalesced on both sides; delta changes only every 59049 elements).
//
// MLP: one outstanding load per wave (~128B / ~800cyc = 0.16 B/cyc) cannot
// feed ~9 B/cyc/SIMD; issue 4 independent transfers per wave before waiting.
//
// CDNA5 path (verified lowering in round 4): async global->LDS->global
// copies tracked by ASYNCcnt; data never touches VGPRs. Double-buffered LDS
// staging + one s_wait_asynccnt(0) per group gives correct ordering. cpol=1
// (TH_NT) on both directions: 344 MB stream, zero reuse, > 192 MB L2.

#define RIGHT     59049u     // 3^10
#define DELTA_I0  118098     // +2*RIGHT (i==0 reads block i=2)
#define DELTA_I12 (-59049)   // -RIGHT   (i==1,2 read block i-1)
#define ELEMS     4
#define BLOCK     256
#define CPOL_NT   1          // gfx12-class CPol: TH[2:0]=1 -> non-temporal

#if defined(__HIP_DEVICE_COMPILE__) &&                                   \
    __has_builtin(__builtin_amdgcn_global_load_async_to_lds_b32) &&      \
    __has_builtin(__builtin_amdgcn_global_store_async_from_lds_b32) &&   \
    __has_builtin(__builtin_amdgcn_s_wait_asynccnt)
#define HAVE_ASYNC_LDS 1
#else
#define HAVE_ASYNC_LDS 0
#endif

#if HAVE_ASYNC_LDS
typedef __attribute__((address_space(1))) int gint_t;   // global (AS1) int
typedef __attribute__((address_space(3))) int lint_t;   // LDS (AS3) int
#endif

__device__ __forceinline__ unsigned xgate_src(unsigned e) {
    unsigned q = e / RIGHT;            // mulhi sequence
    unsigned i = q - (q / 3u) * 3u;    // q % 3
    int      d = (i == 0u) ? DELTA_I0 : DELTA_I12;
    return (unsigned)((int)e + d);     // always in [0, n) by block structure
}

__global__ __launch_bounds__(BLOCK) void xgate_shift_kernel(
    const float* __restrict__ x, float* __restrict__ y, unsigned n)
{
    const unsigned tid  = threadIdx.x;
    const unsigned S    = blockDim.x * gridDim.x;   // element stride
    unsigned       base = blockIdx.x * blockDim.x + tid;

#if HAVE_ASYNC_LDS
    // Ordering per group g (each lane owns private LDS slots, no barriers):
    //   issue 4 async loads into buffer p        (ASYNCcnt += 4)
    //   s_wait_asynccnt 0   -> loads of g done AND stores of g-1 done
    //   issue 4 async stores from buffer p
    //   p ^= 1              -> loads of g+1 use the other buffer (no race
    //                          with in-flight stores of g); buffer p is
    //                          reused only by loads of g+2, issued after the
    //                          wait that retired the stores of g.
    __shared__ float stage[2][ELEMS][BLOCK];
    unsigned p = 0;
    for (; base < n; base += ELEMS * S) {
#pragma unroll
        for (int k = 0; k < ELEMS; ++k) {
            unsigned e = base + (unsigned)k * S;
            if (e < n) {
                unsigned src = xgate_src(e);
                __builtin_amdgcn_global_load_async_to_lds_b32(
                    (gint_t*)(x + src),
                    (lint_t*)&stage[p][k][tid],
                    0, CPOL_NT);
            }
        }
        __builtin_amdgcn_s_wait_asynccnt(0);
#pragma unroll
        for (int k = 0; k < ELEMS; ++k) {
            unsigned e = base + (unsigned)k * S;
            if (e < n) {
                __builtin_amdgcn_global_store_async_from_lds_b32(
                    (gint_t*)(y + e),
                    (lint_t*)&stage[p][k][tid],
                    0, CPOL_NT);
            }
        }
        p ^= 1u;
    }
    // s_endpgm performs an implicit wait-idle, retiring the final stores.
#else
    // Fallback: direct non-temporal copy with 4-deep MLP + gfx1250 L2
    // prefetch of the read stream.
    for (; base < n; base += ELEMS * S) {
        float v[ELEMS];
#pragma unroll
        for (int k = 0; k < ELEMS; ++k) {
            unsigned e = base + (unsigned)k * S;
            if (e < n) {
                unsigned src = xgate_src(e);
                unsigned pf  = src + 4096u;          // 16 KB ahead
                if (pf < n) __builtin_prefetch(x + pf, 0, 3);
                v[k] = __builtin_nontemporal_load(x + src);
            }
        }
#pragma unroll
        for (int k = 0; k < ELEMS; ++k) {
            unsigned e = base + (unsigned)k * S;
            if (e < n) __builtin_nontemporal_store(v[k], y + e);
        }
    }
#endif
}

extern "C" void kernel_launch(void* const* d_in, const int* in_sizes, int n_in,
                              void* d_out, int out_size, void* d_ws, size_t ws_size,
                              hipStream_t stream) {
    (void)in_sizes; (void)n_in; (void)d_ws; (void)ws_size;

    const float* x = (const float*)d_in[0];   // state vector, 3^16 f32
    // d_in[1] is M (3x3) = fixed shift permutation, folded into the address
    // map above (exact in f32).
    float* y = (float*)d_out;

    unsigned n = (unsigned)out_size;                      // 43,046,721
    unsigned blocks = (n + BLOCK * ELEMS - 1u) / (BLOCK * ELEMS);  // one group
    hipLaunchKernelGGL(xgate_shift_kernel, dim3(blocks), dim3(BLOCK), 0, stream,
                       x, y, n);
}